// AlltagCopyCtxGenerator_69801808495267
// MI455X (gfx1250) — compile-verified
//
#include <hip/hip_runtime.h>
#include <hip/hip_bf16.h>

// Problem constants (match reference)
#define VSZ   50000
#define DIM   300
#define DPAD  304          // DIM padded to multiple of 16 for WMMA N-tiles
#define PEXP  8
#define MCAND 512
#define BB    8
#define LL    1024
#define HSF   512
#define TT    (BB*LL)      // 8192
#define EPSV  1e-12f
#define SZS   516          // LDS row stride (pad 512 -> 516: 4-bank rotation, 8B aligned)

typedef float v2f __attribute__((ext_vector_type(2)));
typedef float v8f __attribute__((ext_vector_type(8)));

// ---------------------------------------------------------------- routing ---
__global__ void k_init(int* counts) {
    if (threadIdx.x < PEXP) counts[threadIdx.x] = 0;
}

__global__ void k_route(const int* __restrict__ word, const int* __restrict__ pos,
                        int* __restrict__ counts, int* __restrict__ bucket,
                        float* __restrict__ out_word, float* __restrict__ out_mask,
                        float* __restrict__ out_nn)
{
    int t = blockIdx.x * blockDim.x + threadIdx.x;
    if (t >= TT) return;
    int p = pos[t];
    bool msk = (p < PEXP);
    out_mask[t] = msk ? 1.0f : 0.0f;
    out_nn[t]   = (p < 2) ? 1.0f : 0.0f;
    if (msk) {
        int slot = atomicAdd(&counts[p], 1);
        bucket[p * TT + slot] = t;
    } else {
        out_word[t] = (float)word[t];
    }
}

// Copy base embedding rows for tokens that are NOT routed to any expert.
__global__ void k_base_emb(const int* __restrict__ word, const int* __restrict__ pos,
                           const float* __restrict__ emb, float* __restrict__ out_emb)
{
    int t = blockIdx.x;
    if (pos[t] < PEXP) return;
    const float* src = emb + (size_t)word[t] * DIM;
    float*       dst = out_emb + (size_t)t * DIM;
    for (int d = threadIdx.x; d < DIM; d += blockDim.x) dst[d] = src[d];
}

// Tiled transpose: dec_W[p][k][n] -> Wt[p][n][k]  (WMMA-friendly B layout)
__global__ __launch_bounds__(256) void k_wt(const float* __restrict__ W,
                                            float* __restrict__ Wt)
{
    __shared__ float tile[32][33];
    const int p  = blockIdx.z;
    const int k0 = blockIdx.x * 32;
    const int n0 = blockIdx.y * 32;
    const int tx = threadIdx.x;        // 0..31
    const int ty = threadIdx.y;        // 0..7
    const float* Wp  = W  + (size_t)p * HSF * MCAND;
    float*       Wtp = Wt + (size_t)p * MCAND * HSF;
    #pragma unroll
    for (int j = 0; j < 32; j += 8)
        tile[ty + j][tx] = Wp[(size_t)(k0 + ty + j) * MCAND + (n0 + tx)];
    __syncthreads();
    #pragma unroll
    for (int j = 0; j < 32; j += 8)
        Wtp[(size_t)(n0 + ty + j) * HSF + (k0 + tx)] = tile[tx][ty + j];
}

// Et[p][d][m] = word_emb_weight[tgt_words[p][m]][d]  (transposed, d in [0,DPAD))
__global__ void k_tgt_gather(const int* __restrict__ tgt_words,
                             const float* __restrict__ emb,
                             float* __restrict__ Et)
{
    size_t i = (size_t)blockIdx.x * blockDim.x + threadIdx.x;
    const size_t total = (size_t)PEXP * DPAD * MCAND;
    if (i >= total) return;
    int m  = (int)(i % MCAND);
    int pd = (int)(i / MCAND);
    int d  = pd % DPAD;
    int p  = pd / DPAD;
    Et[i] = (d < DIM) ? emb[(size_t)tgt_words[p * MCAND + m] * DIM + d] : 0.0f;
}

// ------------------------------------------------------------ expert decode ---
// One block = one expert x one 16-token tile. 128 threads = 4 wave32 waves.
__global__ __launch_bounds__(128) void k_decode(
    const float* __restrict__ ctx,        // [T, HS]
    const float* __restrict__ Wt,         // [P, M, HS]   (ws, transposed)
    const float* __restrict__ dec_b,      // [P, M]
    const int*   __restrict__ tgt_words,  // [P, M]
    const float* __restrict__ gumbel,     // [P, T, M]
    const float* __restrict__ Et,         // [P, DPAD, M] (ws, transposed)
    const int*   __restrict__ bucket,     // [P, T]
    const int*   __restrict__ counts,     // [P]
    float* __restrict__ out_word,         // [T]
    float* __restrict__ out_emb)          // [T, DIM]
{
    __shared__ float s_z[16 * SZS];       // logits -> z -> sft (~32.25 KB)
    __shared__ int   s_tok[16];

    const int p      = blockIdx.x >> 9;          // /512
    const int tile   = blockIdx.x & 511;
    const int cnt    = counts[p];
    const int ntiles = (cnt + 15) >> 4;
    if (tile >= ntiles) return;
    const int base   = tile * 16;
    const int nvalid = min(16, cnt - base);

    const int tid  = threadIdx.x;
    const int wave = tid >> 5;
    const int lane = tid & 31;
    const int l16  = lane & 15;
    const int lh   = lane >> 4;                  // 0: lanes 0-15, 1: lanes 16-31
    const int kofs = 2 * lh;                     // lane-half K offset in fragments

    if (tid < 16)
        s_tok[tid] = (tid < nvalid) ? bucket[p * TT + base + tid] : 0;
    __syncthreads();

    // Rows M >= nvalid carry token-0 garbage; their C rows are never stored.
    const float* actx = ctx + (size_t)s_tok[l16] * HSF + kofs;

    // ---- GEMM1: logits[16x512] = ctx_tile * W_p ; 2 acc chains per wave ----
    #pragma unroll
    for (int i = 0; i < 4; ++i) {
        const int nA = (wave + 8 * i) * 16 + l16;
        const int nB = nA + 64;                  // n-tile (wave + 8i + 4)
        const float* WA = Wt + ((size_t)p * MCAND + nA) * HSF + kofs;
        const float* WB = Wt + ((size_t)p * MCAND + nB) * HSF + kofs;
        v8f c0 = {}, c1 = {};
        #pragma unroll 8
        for (int k = 0; k < HSF; k += 4) {
            v2f a, b0, b1;
            a.x  = actx[k];  a.y  = actx[k + 1];   // one global_load_b64
            b0.x = WA[k];    b0.y = WA[k + 1];     // one global_load_b64
            b1.x = WB[k];    b1.y = WB[k + 1];     // one global_load_b64
            c0 = __builtin_amdgcn_wmma_f32_16x16x4_f32(false, a, false, b0,
                                                       (short)0, c0, false, false);
            c1 = __builtin_amdgcn_wmma_f32_16x16x4_f32(false, a, false, b1,
                                                       (short)0, c1, false, false);
        }
        const float biasA = dec_b[p * MCAND + nA];
        const float biasB = dec_b[p * MCAND + nB];
        #pragma unroll
        for (int v = 0; v < 8; ++v) {            // C: M = v + 8*lh, N = nA/nB
            s_z[(v + 8 * lh) * SZS + nA] = c0[v] + biasA;
            s_z[(v + 8 * lh) * SZS + nB] = c1[v] + biasB;
        }
    }
    __syncthreads();

    // ---- softmax -> log(p+eps)+gumbel -> argmax -> softmax ; 4 rows/wave ----
    const size_t gbase = (size_t)p * TT * MCAND;
    for (int r = wave * 4; r < wave * 4 + 4; ++r) {
        float*       zr   = s_z + r * SZS;
        const int    trow = s_tok[r];
        const float* grow = gumbel + gbase + (size_t)trow * MCAND;

        float mx = -3.402823466e38f;
        for (int m = lane; m < MCAND; m += 32) mx = fmaxf(mx, zr[m]);
        for (int off = 16; off; off >>= 1) mx = fmaxf(mx, __shfl_xor(mx, off, 32));

        float s = 0.0f;
        for (int m = lane; m < MCAND; m += 32) s += __expf(zr[m] - mx);
        for (int off = 16; off; off >>= 1) s += __shfl_xor(s, off, 32);
        const float inv = 1.0f / s;

        float m2 = -3.402823466e38f; int i2 = MCAND;
        for (int m = lane; m < MCAND; m += 32) {
            float sp = __expf(zr[m] - mx) * inv;
            float z  = __logf(sp + EPSV) + grow[m];
            zr[m] = z;
            if (z > m2) { m2 = z; i2 = m; }
        }
        for (int off = 16; off; off >>= 1) {
            float om = __shfl_xor(m2, off, 32);
            int   oi = __shfl_xor(i2, off, 32);
            if (om > m2 || (om == m2 && oi < i2)) { m2 = om; i2 = oi; }
        }

        float s2 = 0.0f;
        for (int m = lane; m < MCAND; m += 32) s2 += __expf(zr[m] - m2);
        for (int off = 16; off; off >>= 1) s2 += __shfl_xor(s2, off, 32);
        const float inv2 = 1.0f / s2;

        for (int m = lane; m < MCAND; m += 32) zr[m] = __expf(zr[m] - m2) * inv2;

        if (lane == 0 && r < nvalid)
            out_word[trow] = (float)tgt_words[p * MCAND + i2];
    }
    __syncthreads();

    // ---- GEMM2: x_emb[16x300] = sft[16x512] * tgt_emb_p[512xDPAD] ----
    const float* sfrow = s_z + l16 * SZS + kofs;          // A rows from LDS
    for (int nt = wave; nt < DPAD / 16; nt += 4) {        // 19 N-tiles
        const int n = nt * 16 + l16;
        const float* Ep = Et + ((size_t)p * DPAD + n) * MCAND + kofs;
        v8f c = {};
        #pragma unroll 8
        for (int k = 0; k < MCAND; k += 4) {
            v2f a, b;
            a.x = sfrow[k]; a.y = sfrow[k + 1];           // one ds_load_b64
            b.x = Ep[k];    b.y = Ep[k + 1];              // one global_load_b64
            c = __builtin_amdgcn_wmma_f32_16x16x4_f32(false, a, false, b,
                                                      (short)0, c, false, false);
        }
        if (n < DIM) {
            #pragma unroll
            for (int v = 0; v < 8; ++v) {
                const int mr = v + 8 * lh;
                if (mr < nvalid)
                    out_emb[(size_t)s_tok[mr] * DIM + n] = c[v];
            }
        }
    }
}

// ------------------------------------------------------------------- launch ---
extern "C" void kernel_launch(void* const* d_in, const int* in_sizes, int n_in,
                              void* d_out, int out_size, void* d_ws, size_t ws_size,
                              hipStream_t stream) {
    (void)in_sizes; (void)n_in; (void)out_size; (void)ws_size;

    const int*   inp_word  = (const int*)  d_in[0];   // [B,L]
    const int*   inp_pos   = (const int*)  d_in[1];   // [B,L]
    const float* ctx       = (const float*)d_in[2];   // [B,L,HS]
    const float* word_emb  = (const float*)d_in[3];   // [V,D]
    const float* dec_W     = (const float*)d_in[4];   // [P,HS,M]
    const float* dec_b     = (const float*)d_in[5];   // [P,M]
    const int*   tgt_words = (const int*)  d_in[6];   // [P,M]
    const float* gumbel    = (const float*)d_in[7];   // [P,B*L,M]

    // Output layout (flattened tuple, all float):
    float* out_word = (float*)d_out;                       // [T]
    float* out_emb  = out_word + TT;                       // [T,DIM]
    float* out_mask = out_emb + (size_t)TT * DIM;          // [T]
    float* out_nn   = out_mask + TT;                       // [T]

    // Workspace layout (~13.5 MB):
    char*  ws     = (char*)d_ws;
    int*   counts = (int*)ws;                                          // P ints
    int*   bucket = (int*)(ws + 256);                                  // P*T ints
    float* WtW    = (float*)(ws + 256 + (size_t)PEXP * TT * 4);        // P*M*HS f32
    float* EtW    = WtW + (size_t)PEXP * MCAND * HSF;                  // P*DPAD*M f32

    k_init<<<1, 32, 0, stream>>>(counts);
    k_route<<<TT / 256, 256, 0, stream>>>(inp_word, inp_pos, counts, bucket,
                                          out_word, out_mask, out_nn);
    k_base_emb<<<TT, 128, 0, stream>>>(inp_word, inp_pos, word_emb, out_emb);

    k_wt<<<dim3(HSF / 32, MCAND / 32, PEXP), dim3(32, 8), 0, stream>>>(dec_W, WtW);

    const size_t tg_total = (size_t)PEXP * DPAD * MCAND;
    k_tgt_gather<<<(unsigned)((tg_total + 255) / 256), 256, 0, stream>>>(
        tgt_words, word_emb, EtW);

    k_decode<<<PEXP * 512, 128, 0, stream>>>(ctx, WtW, dec_b, tgt_words, gumbel,
                                             EtW, bucket, counts,
                                             out_word, out_emb);
}